// PaperArchitecture_12249246728616
// MI455X (gfx1250) — compile-verified
//
#include <hip/hip_runtime.h>
#include <math.h>

// ---------------------------------------------------------------------------
// MI455X (gfx1250) implementation.
//
// Input pointer layout (jax tree_leaves flattening: dicts sorted by key,
// lists/tuples in order), top-level dict insertion order x, params, S:
//   0: x [64,256,3,11,11] f32
//   params leaves (1..50):
//     block k (k=0..2), base = 1+12k, leaves sorted b1,b2,bn1(g,be,m,v),
//       bn2(g,be,m,v),w1,w2:
//       +0 b1, +1 b2, +2..5 bn1, +6..9 bn2, +10 w1, +11 w2
//     37 cmp_b, 38..41 cmp_bn(g,be,m,v), 42 cmp_w, 43 fc_b, 44 fc_w,
//     45..50 gcn [(W0,b0),(W1,b1),(W2,b2)]
//   51: S [64,256,256] i32
// ---------------------------------------------------------------------------

typedef float v2f __attribute__((ext_vector_type(2)));
typedef float v8f __attribute__((ext_vector_type(8)));

#define BN_EPS 1e-5f
#define NIMG   16384     // B*N
#define NB     64
#define NN     256
#define FD     128

// ------------------------- CNN feature extractor ---------------------------

struct ConvP {
  const float* w; const float* b;
  const float* g; const float* be; const float* m; const float* v;
};

struct CnnArgs {
  const float* x;      // [16384,3,11,11]
  ConvP c[6];          // conv1..conv6 (block-major)
  ConvP cmp;           // 1x1 compress conv + bn
  float* feat;         // [16384,128] out
};

__device__ __forceinline__ void prep_affine(float* sc, float* sh, const ConvP& p,
                                            int c, int tid) {
  // fold conv-bias + BN into  out = acc*sc[c] + sh[c]
  for (int i = tid; i < c; i += 128) {
    float s = p.g[i] * rsqrtf(p.v[i] + BN_EPS);
    sc[i] = s;
    sh[i] = (p.b[i] - p.m[i]) * s + p.be[i];
  }
}

__device__ __forceinline__ void conv3x3_bn_relu(const float* in, float* out,
                                                int cin, int cout, int H, int W,
                                                const float* __restrict__ wgt,
                                                const float* sc, const float* sh,
                                                int tid) {
  const int HW = H * W;
  for (int idx = tid; idx < cout * HW; idx += 128) {
    const int co = idx / HW;
    const int p  = idx - co * HW;
    const int oy = p / W, ox = p - (p / W) * W;
    float acc = 0.f;
    const float* wk = wgt + (size_t)co * cin * 9;
    for (int ci = 0; ci < cin; ++ci) {
      const float* ip = in + ci * HW;
      const float* wc = wk + ci * 9;
#pragma unroll
      for (int ky = 0; ky < 3; ++ky) {
        const int iy = oy + ky - 1;
        if (iy < 0 || iy >= H) continue;
#pragma unroll
        for (int kx = 0; kx < 3; ++kx) {
          const int ix = ox + kx - 1;
          if (ix < 0 || ix >= W) continue;
          acc = fmaf(ip[iy * W + ix], wc[ky * 3 + kx], acc);
        }
      }
    }
    out[idx] = fmaxf(acc * sc[co] + sh[co], 0.f);
  }
}

__device__ __forceinline__ void maxpool2(const float* in, float* out,
                                         int c, int H, int W, int tid) {
  const int oh = H >> 1, ow = W >> 1;
  for (int idx = tid; idx < c * oh * ow; idx += 128) {
    const int cc = idx / (oh * ow);
    const int p  = idx - cc * (oh * ow);
    const int oy = p / ow, ox = p - (p / ow) * ow;
    const float* ip = in + cc * H * W + (2 * oy) * W + 2 * ox;
    out[idx] = fmaxf(fmaxf(ip[0], ip[1]), fmaxf(ip[W], ip[W + 1]));
  }
}

__global__ __launch_bounds__(128) void cnn_features_kernel(CnnArgs A) {
  __shared__ float buf0[1600];   // input(363)/pool1(800)/conv3(1600)/conv4(256)/pool3(128)
  __shared__ float buf1[3872];   // conv1(3872)/conv2(800)/pool2(256)/conv5(512)/conv6(128)
  __shared__ float sc[128];
  __shared__ float sh[128];
  const int tid = threadIdx.x;
  const int img = blockIdx.x;

  const float* xin = A.x + (size_t)img * 363;
  for (int i = tid; i < 363; i += 128) buf0[i] = xin[i];
  prep_affine(sc, sh, A.c[0], 32, tid);
  __syncthreads();

  conv3x3_bn_relu(buf0, buf1, 3, 32, 11, 11, A.c[0].w, sc, sh, tid);   // 32x11x11
  __syncthreads();
  maxpool2(buf1, buf0, 32, 11, 11, tid);                               // 32x5x5
  prep_affine(sc, sh, A.c[1], 32, tid);
  __syncthreads();
  conv3x3_bn_relu(buf0, buf1, 32, 32, 5, 5, A.c[1].w, sc, sh, tid);    // 32x5x5
  __syncthreads();
  prep_affine(sc, sh, A.c[2], 64, tid);
  __syncthreads();
  conv3x3_bn_relu(buf1, buf0, 32, 64, 5, 5, A.c[2].w, sc, sh, tid);    // 64x5x5
  __syncthreads();
  maxpool2(buf0, buf1, 64, 5, 5, tid);                                 // 64x2x2
  prep_affine(sc, sh, A.c[3], 64, tid);
  __syncthreads();
  conv3x3_bn_relu(buf1, buf0, 64, 64, 2, 2, A.c[3].w, sc, sh, tid);    // 64x2x2
  __syncthreads();
  prep_affine(sc, sh, A.c[4], 128, tid);
  __syncthreads();
  conv3x3_bn_relu(buf0, buf1, 64, 128, 2, 2, A.c[4].w, sc, sh, tid);   // 128x2x2
  __syncthreads();
  maxpool2(buf1, buf0, 128, 2, 2, tid);                                // 128x1x1
  prep_affine(sc, sh, A.c[5], 128, tid);
  __syncthreads();

  // conv6: 3x3 pad=1 on 1x1 spatial => only the center tap contributes.
  {
    const int o = tid;  // 128 threads exactly
    const float* wr = A.c[5].w + (size_t)o * 128 * 9 + 4;
    float acc = 0.f;
    for (int i = 0; i < 128; ++i) acc = fmaf(buf0[i], wr[(size_t)i * 9], acc);
    buf1[o] = fmaxf(acc * sc[o] + sh[o], 0.f);
  }
  __syncthreads();
  prep_affine(sc, sh, A.cmp, 128, tid);
  __syncthreads();

  // global mean over 1x1 == identity; compress 1x1 conv + BN + ReLU
  {
    const int o = tid;
    const float* wr = A.cmp.w + (size_t)o * 128;
    float acc = 0.f;
    for (int i = 0; i < 128; ++i) acc = fmaf(buf1[i], wr[i], acc);
    A.feat[(size_t)img * 128 + o] = fmaxf(acc * sc[o] + sh[o], 0.f);
  }
}

// --------------------------- GCN normalization -----------------------------

__global__ __launch_bounds__(256) void dinv_kernel(const int* __restrict__ S,
                                                   float* __restrict__ dinv) {
  const int i = blockIdx.x * 256 + threadIdx.x;   // b*256+node
  const int* row = S + (size_t)i * NN;
  int s = 0;
  for (int j = 0; j < NN; ++j) s += row[j];
  s -= row[i & (NN - 1)];                         // drop diagonal (S*(1-I))
  dinv[i] = rsqrtf((float)s + 2.0f);              // +2 self-loop (improved=True)
}

__global__ __launch_bounds__(256) void norm_kernel(const int* __restrict__ S,
                                                   const float* __restrict__ dinv,
                                                   float* __restrict__ nrm) {
  const int bi = blockIdx.x;                      // b*256+i
  const int j  = threadIdx.x;
  const int b  = bi >> 8, i = bi & (NN - 1);
  const float a = (i == j) ? 2.0f : (float)S[(size_t)bi * NN + j];
  nrm[(size_t)bi * NN + j] = dinv[bi] * a * dinv[b * NN + j];
}

// --------------------- fp32 WMMA 16x64 GEMM tiles --------------------------
// V_WMMA_F32_16X16X4_F32 operand layouts (ISA 7.12.2):
//   A 16x4: lane = m | khalf; VGPR0/1 = consecutive K within lane.
//   B 4x16: lane = n | khalf; VGPR0/1 = consecutive K rows.
//   C/D:    VGPR r holds rows r (lanes 0-15) and r+8 (lanes 16-31).
// Each wave owns a 16x64 tile: one A fetch feeds 4 independent WMMA chains
// (4-way ILP on the matrix pipe; breaks the C->D accumulation latency chain)
// and cuts A-panel traffic 4x versus a 16x16 tile.

struct Acc4 { v8f a0, a1, a2, a3; };

__device__ __forceinline__ Acc4 wmma_f32_tile4(const float* __restrict__ Ap, int lda,
                                               const float* __restrict__ Bp, int ldb,
                                               int K, int lane) {
  const int m  = lane & 15;
  const int kh = (lane >> 4) << 1;   // 0 or 2
  Acc4 acc; acc.a0 = {}; acc.a1 = {}; acc.a2 = {}; acc.a3 = {};
  for (int k = 0; k < K; k += 4) {
    const float* ap = Ap + (size_t)m * lda + k + kh;
    v2f av; av.x = ap[0]; av.y = ap[1];
    const float* bp = Bp + (size_t)(k + kh) * ldb + m;
    v2f b0; b0.x = bp[0];  b0.y = bp[ldb];
    v2f b1; b1.x = bp[16]; b1.y = bp[ldb + 16];
    v2f b2; b2.x = bp[32]; b2.y = bp[ldb + 32];
    v2f b3; b3.x = bp[48]; b3.y = bp[ldb + 48];
    acc.a0 = __builtin_amdgcn_wmma_f32_16x16x4_f32(false, av, false, b0,
                                                   (short)0, acc.a0, false, false);
    acc.a1 = __builtin_amdgcn_wmma_f32_16x16x4_f32(false, av, false, b1,
                                                   (short)0, acc.a1, false, false);
    acc.a2 = __builtin_amdgcn_wmma_f32_16x16x4_f32(false, av, false, b2,
                                                   (short)0, acc.a2, false, false);
    acc.a3 = __builtin_amdgcn_wmma_f32_16x16x4_f32(false, av, false, b3,
                                                   (short)0, acc.a3, false, false);
  }
  return acc;
}

// tmp[16384x128] = H[16384x128] @ W[128x128]   (16x64 tile per wave)
__global__ __launch_bounds__(32) void gemm_hw_kernel(const float* __restrict__ H,
                                                     const float* __restrict__ Wg,
                                                     float* __restrict__ out) {
  const int lane = threadIdx.x;
  const int mt  = blockIdx.x;          // 0..1023
  const int nt0 = blockIdx.y * 64;     // 0 or 64
  Acc4 acc = wmma_f32_tile4(H + (size_t)mt * 16 * FD, FD, Wg + nt0, FD, FD, lane);
  const int n  = lane & 15;
  const int mh = (lane >> 4) * 8;
  float* op = out + ((size_t)mt * 16 + mh) * FD + nt0 + n;
#pragma unroll
  for (int r = 0; r < 8; ++r) {
    op[(size_t)r * FD]      = acc.a0[r];
    op[(size_t)r * FD + 16] = acc.a1[r];
    op[(size_t)r * FD + 32] = acc.a2[r];
    op[(size_t)r * FD + 48] = acc.a3[r];
  }
}

// out[b] = relu(norm[b](256x256) @ T[b](256x128) + bias)   (16x64 per wave)
__global__ __launch_bounds__(32) void gcn_agg_kernel(const float* __restrict__ Nrm,
                                                     const float* __restrict__ T,
                                                     const float* __restrict__ bias,
                                                     float* __restrict__ out) {
  const int lane = threadIdx.x;
  const int mt  = blockIdx.x;          // 0..15
  const int nt0 = blockIdx.y * 64;     // 0 or 64
  const int b   = blockIdx.z;          // 0..63
  Acc4 acc = wmma_f32_tile4(Nrm + (size_t)b * NN * NN + (size_t)mt * 16 * NN, NN,
                            T + (size_t)b * NN * FD + nt0, FD, NN, lane);
  const int n  = lane & 15;
  const int mh = (lane >> 4) * 8;
  const float bv0 = bias[nt0 + n];
  const float bv1 = bias[nt0 + n + 16];
  const float bv2 = bias[nt0 + n + 32];
  const float bv3 = bias[nt0 + n + 48];
  float* op = out + (size_t)b * NN * FD + ((size_t)mt * 16 + mh) * FD + nt0 + n;
#pragma unroll
  for (int r = 0; r < 8; ++r) {
    op[(size_t)r * FD]      = fmaxf(acc.a0[r] + bv0, 0.f);
    op[(size_t)r * FD + 16] = fmaxf(acc.a1[r] + bv1, 0.f);
    op[(size_t)r * FD + 32] = fmaxf(acc.a2[r] + bv2, 0.f);
    op[(size_t)r * FD + 48] = fmaxf(acc.a3[r] + bv3, 0.f);
  }
}

// ------------------------------- final FC ----------------------------------

__global__ __launch_bounds__(256) void fc_kernel(const float* __restrict__ h,
                                                 const float* __restrict__ Wf,
                                                 const float* __restrict__ bf,
                                                 float* __restrict__ out) {
  const int idx = blockIdx.x * 256 + threadIdx.x;
  if (idx >= NIMG * 5) return;
  const int row = idx / 5, c = idx - row * 5;
  const float* hr = h + (size_t)row * FD;
  float acc = bf[c];
  for (int i = 0; i < FD; ++i) acc = fmaf(hr[i], Wf[i * 5 + c], acc);
  out[idx] = acc;
}

// ------------------------------- launcher ----------------------------------

extern "C" void kernel_launch(void* const* d_in, const int* in_sizes, int n_in,
                              void* d_out, int out_size, void* d_ws, size_t ws_size,
                              hipStream_t stream) {
  (void)in_sizes; (void)n_in; (void)out_size; (void)ws_size;
  const float* x = (const float*)d_in[0];
  const int*   S = (const int*)d_in[51];

  float* ws    = (float*)d_ws;
  float* feat  = ws;                 // 16384*128
  float* feat2 = ws + 2097152;       // 16384*128
  float* gtmp  = ws + 4194304;       // 16384*128
  float* nrm   = ws + 6291456;       // 64*256*256
  float* dinv  = ws + 10485760;      // 16384

  CnnArgs ca;
  ca.x = x; ca.feat = feat;
  const int wIdx[6]  = {11, 12, 23, 24, 35, 36};
  const int bIdx[6]  = { 1,  2, 13, 14, 25, 26};
  const int bnIdx[6] = { 3,  7, 15, 19, 27, 31};  // g,be,m,v consecutive
  for (int k = 0; k < 6; ++k) {
    ca.c[k].w  = (const float*)d_in[wIdx[k]];
    ca.c[k].b  = (const float*)d_in[bIdx[k]];
    ca.c[k].g  = (const float*)d_in[bnIdx[k] + 0];
    ca.c[k].be = (const float*)d_in[bnIdx[k] + 1];
    ca.c[k].m  = (const float*)d_in[bnIdx[k] + 2];
    ca.c[k].v  = (const float*)d_in[bnIdx[k] + 3];
  }
  ca.cmp.w  = (const float*)d_in[42];
  ca.cmp.b  = (const float*)d_in[37];
  ca.cmp.g  = (const float*)d_in[38];
  ca.cmp.be = (const float*)d_in[39];
  ca.cmp.m  = (const float*)d_in[40];
  ca.cmp.v  = (const float*)d_in[41];

  cnn_features_kernel<<<NIMG, 128, 0, stream>>>(ca);
  dinv_kernel<<<NB, 256, 0, stream>>>(S, dinv);
  norm_kernel<<<NB * NN, 256, 0, stream>>>(S, dinv, nrm);

  float* cur = feat;
  float* nxt = feat2;
  for (int l = 0; l < 3; ++l) {
    gemm_hw_kernel<<<dim3(NIMG / 16, FD / 64), 32, 0, stream>>>(
        cur, (const float*)d_in[45 + 2 * l], gtmp);
    gcn_agg_kernel<<<dim3(NN / 16, FD / 64, NB), 32, 0, stream>>>(
        nrm, gtmp, (const float*)d_in[46 + 2 * l], nxt);
    float* t = cur; cur = nxt; nxt = t;
  }

  fc_kernel<<<(NIMG * 5 + 255) / 256, 256, 0, stream>>>(
      cur, (const float*)d_in[44], (const float*)d_in[43], (float*)d_out);
}